// DetectPeaks_46720654246500
// MI455X (gfx1250) — compile-verified
//
#include <hip/hip_runtime.h>

#define W_LAG 8192
#define NROWS 4096
#define TPB   256
#define EPT   32      // W_LAG / TPB
#define NLAG  4096    // W_LAG / 2
#define NWAVES (TPB / 32)

// Branchless insert of candidate (cs, ci) into a descending top-3 list with
// stable tie-break (equal score -> lower index first), matching lax.top_k.
// List entries are strictly ordered under (score desc, index asc), so
// b0 => b1 => b2, which the select chain below relies on.
__device__ __forceinline__ void topk3_insert(float cs, int ci, float s[3], int id[3]) {
    bool b0 = (cs > s[0]) || ((cs == s[0]) && (ci < id[0]));
    bool b1 = (cs > s[1]) || ((cs == s[1]) && (ci < id[1]));
    bool b2 = (cs > s[2]) || ((cs == s[2]) && (ci < id[2]));
    float ns0 = b0 ? cs   : s[0];
    int   ni0 = b0 ? ci   : id[0];
    float ns1 = b0 ? s[0] : (b1 ? cs : s[1]);
    int   ni1 = b0 ? id[0] : (b1 ? ci : id[1]);
    float ns2 = b1 ? s[1] : (b2 ? cs : s[2]);
    int   ni2 = b1 ? id[1] : (b2 ? ci : id[2]);
    s[0] = ns0; id[0] = ni0;
    s[1] = ns1; id[1] = ni1;
    s[2] = ns2; id[2] = ni2;
}

__global__ __launch_bounds__(TPB) void detect_peaks_kernel(
    const float* __restrict__ xcorr, float* __restrict__ out) {
    // Row data lives at buf[4 .. 4+W_LAG) (byte offset 16 -> every lane's B128
    // async-LDS destination is 16B aligned). -inf halos at buf[3] / buf[4+W_LAG].
    __shared__ __align__(16) float buf[W_LAG + 8];
    __shared__ float rs[NWAVES * 3];
    __shared__ int   ri[NWAVES * 3];

    const int t   = threadIdx.x;
    const int row = blockIdx.x;
    const float* rowp = xcorr + (size_t)row * W_LAG;

    // Low 32 bits of a generic pointer to LDS == wave-relative LDS byte address.
    unsigned ldsBase = (unsigned)(unsigned long long)(&buf[4]);
    unsigned byteOff = (unsigned)t * 16u;

    // Stream the whole 32KiB row into LDS with the CDNA5 async copy engine.
    // Non-temporal: the stream (128MiB total) has no reuse, keep it out of L2.
#pragma unroll
    for (int sweep = 0; sweep < 8; ++sweep) {
        unsigned off   = byteOff + (unsigned)(sweep * (TPB * 16));
        unsigned laddr = ldsBase + off;
        asm volatile("global_load_async_to_lds_b128 %0, %1, %2 th:TH_LOAD_NT"
                     :
                     : "v"(laddr), "v"(off), "s"(rowp)
                     : "memory");
    }

    if (t == 0) {
        buf[3]         = -__builtin_inff();   // left  -inf pad (max-pool padding)
        buf[4 + W_LAG] = -__builtin_inff();   // right -inf pad
    }

    // Wait for this wave's async transfers, then cross-wave visibility.
    asm volatile("s_wait_asynccnt 0" ::: "memory");
    __syncthreads();

    const float* bufd = &buf[4];

    float s[3]  = {-__builtin_inff(), -__builtin_inff(), -__builtin_inff()};
    int   id[3] = {0x7fffffff, 0x7fffffff, 0x7fffffff};

    // Strided scan: consecutive lanes hit consecutive LDS banks (conflict-free).
    // score = x * keep  ==  (x >= left && x >= right) ? x : 0
#pragma unroll 8
    for (int e = 0; e < EPT; ++e) {
        int   ix = e * TPB + t;
        float xl = bufd[ix - 1];
        float xc = bufd[ix];
        float xr = bufd[ix + 1];
        float cs = ((xc >= xl) && (xc >= xr)) ? xc : 0.0f;
        topk3_insert(cs, ix, s, id);
    }

    // Intra-wave (wave32) reduction via shuffles: 5 steps, no barriers.
#pragma unroll
    for (int off = 16; off >= 1; off >>= 1) {
        float os0 = __shfl_down(s[0], off);
        float os1 = __shfl_down(s[1], off);
        float os2 = __shfl_down(s[2], off);
        int   oi0 = __shfl_down(id[0], off);
        int   oi1 = __shfl_down(id[1], off);
        int   oi2 = __shfl_down(id[2], off);
        topk3_insert(os0, oi0, s, id);
        topk3_insert(os1, oi1, s, id);
        topk3_insert(os2, oi2, s, id);
    }

    const int wave = t >> 5;
    const int lane = t & 31;
    if (lane == 0) {
        rs[wave * 3 + 0] = s[0];  rs[wave * 3 + 1] = s[1];  rs[wave * 3 + 2] = s[2];
        ri[wave * 3 + 0] = id[0]; ri[wave * 3 + 1] = id[1]; ri[wave * 3 + 2] = id[2];
    }
    __syncthreads();

    if (t == 0) {
        // Merge the other 7 wave lists (wave 0's result is already in s/id).
#pragma unroll
        for (int w = 1; w < NWAVES; ++w) {
            topk3_insert(rs[w * 3 + 0], ri[w * 3 + 0], s, id);
            topk3_insert(rs[w * 3 + 1], ri[w * 3 + 1], s, id);
            topk3_insert(rs[w * 3 + 2], ri[w * 3 + 2], s, id);
        }
        int top = id[0];
#pragma unroll
        for (int j = 0; j < 3; ++j) {
            int nb = top - 1 + j;
            nb = nb < 0 ? 0 : (nb > W_LAG - 1 ? W_LAG - 1 : nb);
            // Output 0: neighbor_score [4096,3]
            out[(size_t)row * 3 + j] = bufd[nb];
            // Output 1: topk_scores [4096,3]
            out[(size_t)NROWS * 3 + (size_t)row * 3 + j] = s[j];
            // Output 2: topk_index [4096,3] (= index - nlag), stored into float buffer
            out[(size_t)2 * NROWS * 3 + (size_t)row * 3 + j] = (float)(id[j] - NLAG);
        }
    }
}

extern "C" void kernel_launch(void* const* d_in, const int* in_sizes, int n_in,
                              void* d_out, int out_size, void* d_ws, size_t ws_size,
                              hipStream_t stream) {
    (void)in_sizes; (void)n_in; (void)out_size; (void)d_ws; (void)ws_size;
    const float* x  = (const float*)d_in[0];
    float* out      = (float*)d_out;
    detect_peaks_kernel<<<NROWS, TPB, 0, stream>>>(x, out);
}